// DetTopKLoss_11733850652799
// MI455X (gfx1250) — compile-verified
//
#include <hip/hip_runtime.h>
#include <hip/hip_bf16.h>
#include <stdint.h>

#define BATCH 16
#define HH 1024
#define WW 1024
#define NPIX (HH * WW)
#define NMS_R 7
#define TOPK 2048
#define CAP 8192          // candidate capacity per image (expected ~2330)
#define TILE 32
#define HALO (TILE + 2 * NMS_R)   // 46

// ws layout (bytes)
#define OFF_CANDCNT  0        // int[16]
#define OFF_KCOUNT   64       // int[16]
#define OFF_ACCLOSS  128      // float[16]
#define OFF_ACCMASK  192      // float[16]
#define OFF_ACCDELTA 256      // float[16]
#define OFF_KP       1024     // int[16][2048]
#define OFF_CAND     262144   // u64[16][CAP]

// normalized 1-D Gaussian, ks=7, sigma=1
__device__ __constant__ float GK[7] = {
    0.004433048f, 0.054005582f, 0.242036229f, 0.399050280f,
    0.242036229f, 0.054005582f, 0.004433048f};

typedef __attribute__((ext_vector_type(2))) float v2f;
typedef __attribute__((ext_vector_type(8))) float v8f;

// ---- wave32 sum via V_WMMA_F32_16X16X4_F32 -------------------------------
// A[m][k]: lane j holds A[j%16][2*(j/16)+v] for v=0,1.  Put partial in slot 0.
// B = ones(4x16)  =>  D[m][n] = p_m + p_{m+16} (independent of n).
// D VGPR i on lane L: M = i + 8*(L/16), N = L%16.  Sum 8 regs then xor-16.
__device__ inline float wave_sum_wmma(float x) {
  v2f a; a[0] = x;    a[1] = 0.0f;
  v2f o; o[0] = 1.0f; o[1] = 1.0f;
  v8f c = {};
  c = __builtin_amdgcn_wmma_f32_16x16x4_f32(false, a, false, o, (short)0, c,
                                            false, false);
  float s = 0.0f;
#pragma unroll
  for (int i = 0; i < 8; ++i) s += c[i];
  s += __shfl_xor(s, 16, 32);
  return s;  // all lanes hold total
}

// ---- K0: zero the small accumulator region -------------------------------
__global__ void k_zero(unsigned* ws) {
  if (threadIdx.x < 256) ws[threadIdx.x] = 0u;  // first 1024 bytes
}

// ---- K1: fused 15x15 NMS + mask + candidate compaction -------------------
__global__ void k_nms_collect(const float* __restrict__ score2,
                              const int* __restrict__ mask,
                              int* __restrict__ candCount,
                              unsigned long long* __restrict__ cand) {
  __shared__ float tile[HALO][HALO + 2];
  __shared__ float hmax[HALO][TILE];
  const int b  = blockIdx.z;
  const int bx = blockIdx.x * TILE;
  const int by = blockIdx.y * TILE;
  const int tid = threadIdx.x;
  const float* s2 = score2 + (size_t)b * NPIX;
  const int*   mk = mask   + (size_t)b * NPIX;

  for (int i = tid; i < HALO * HALO; i += blockDim.x) {
    int hy = i / HALO, hx = i % HALO;
    int gy = by + hy - NMS_R, gx = bx + hx - NMS_R;
    float v = -INFINITY;
    if (gy >= 0 && gy < HH && gx >= 0 && gx < WW) v = s2[gy * WW + gx];
    tile[hy][hx] = v;
  }
  __syncthreads();

  for (int i = tid; i < HALO * TILE; i += blockDim.x) {
    int hy = i / TILE, ox = i % TILE;
    float m = tile[hy][ox];
#pragma unroll
    for (int d = 1; d < 15; ++d) m = fmaxf(m, tile[hy][ox + d]);
    hmax[hy][ox] = m;
  }
  __syncthreads();

  for (int i = tid; i < TILE * TILE; i += blockDim.x) {
    int oy = i / TILE, ox = i % TILE;
    float m = hmax[oy][ox];
#pragma unroll
    for (int d = 1; d < 15; ++d) m = fmaxf(m, hmax[oy + d][ox]);
    float cv = tile[oy + NMS_R][ox + NMS_R];
    int gy = by + oy, gx = bx + ox;
    if (cv == m && cv > 0.0f && mk[gy * WW + gx] != 0) {
      int pos = atomicAdd(&candCount[b], 1);
      if (pos < CAP) {
        unsigned vb  = __float_as_uint(cv);
        unsigned idx = (unsigned)(gy * WW + gx);
        cand[(size_t)b * CAP + pos] =
            ((unsigned long long)vb << 32) |
            (unsigned long long)(0xFFFFFFFFu - idx);  // value desc, index asc
      }
    }
  }
}

// ---- K2: per-image exact top-K via bitonic sort in LDS -------------------
__global__ __launch_bounds__(1024) void k_select(
    const int* __restrict__ candCount,
    const unsigned long long* __restrict__ cand,
    int* __restrict__ kcount, int* __restrict__ kp) {
  __shared__ unsigned long long keys[CAP];  // 64 KB (WGP has 320 KB)
  const int b = blockIdx.x;
  const int tid = threadIdx.x;
  int count = candCount[b];
  if (count > CAP) count = CAP;
  for (int i = tid; i < CAP; i += blockDim.x)
    keys[i] = (i < count) ? cand[(size_t)b * CAP + i] : 0ull;
  __syncthreads();

  for (int size = 2; size <= CAP; size <<= 1)
    for (int stride = size >> 1; stride > 0; stride >>= 1) {
      for (int i = tid; i < CAP / 2; i += blockDim.x) {
        int lo = (i / stride) * (stride * 2) + (i % stride);
        int hi = lo + stride;
        bool desc = ((lo & size) == 0);
        unsigned long long a = keys[lo], c = keys[hi];
        bool sw = desc ? (a < c) : (a > c);
        if (sw) { keys[lo] = c; keys[hi] = a; }
      }
      __syncthreads();
    }

  int kc = count < TOPK ? count : TOPK;
  for (int j = tid; j < kc; j += blockDim.x) {
    unsigned idx = 0xFFFFFFFFu - (unsigned)(keys[j] & 0xFFFFFFFFull);
    kp[b * TOPK + j] = (int)idx;
  }
  if (tid == 0) kcount[b] = kc;
}

// ---- K3: sparse Gaussian-stamp correction: sum m*(g^2 - 2*s1*g) ----------
__global__ void k_stamp(const float* __restrict__ score1,
                        const int* __restrict__ mask,
                        const int* __restrict__ kcount,
                        const int* __restrict__ kp,
                        float* __restrict__ accDelta) {
  const int b = blockIdx.x / TOPK;
  const int j = blockIdx.x % TOPK;
  if (j >= kcount[b]) return;
  const int t = threadIdx.x;  // 64 threads, 49 active
  float d = 0.0f;
  if (t < 49) {
    int dy = t / 7 - 3, dx = t % 7 - 3;
    int idx = kp[b * TOPK + j];
    int y = idx / WW + dy, x = idx % WW + dx;
    if (y >= 0 && y < HH && x >= 0 && x < WW) {
      float g = GK[dy + 3] * GK[dx + 3];
      size_t p = (size_t)b * NPIX + (size_t)y * WW + x;
      if (mask[p] != 0) {
        float s = score1[p];
        d = g * g - 2.0f * s * g;   // (s-g)^2 - s^2
      }
    }
  }
#pragma unroll
  for (int off = 16; off > 0; off >>= 1) d += __shfl_xor(d, off, 32);
  __shared__ float part[2];
  if ((t & 31) == 0) part[t >> 5] = d;
  __syncthreads();
  if (t == 0) atomicAdd(&accDelta[b], part[0] + part[1]);
}

// ---- K4: dense base pass: sum m*s1^2 and sum m ---------------------------
// Branchless + b128 vector loads; WMMA wave reduction at the end.
__global__ void k_base(const float* __restrict__ score1,
                       const int* __restrict__ mask,
                       float* __restrict__ accLoss,
                       float* __restrict__ accMask) {
  const int b = blockIdx.y;
  const size_t begin = (size_t)b * NPIX + (size_t)blockIdx.x * 4096;
  const int tid = threadIdx.x;
  const float4* s4 = (const float4*)(score1 + begin);
  const int4*   m4 = (const int4*)(mask + begin);

  // prefetch next block-stride chunk into L2 (global_prefetch_b8)
  __builtin_prefetch(score1 + begin + 4096 + tid * 4, 0, 1);
  __builtin_prefetch(mask   + begin + 4096 + tid * 4, 0, 1);

  float aL = 0.0f, aM = 0.0f;
#pragma unroll
  for (int k = 0; k < 4; ++k) {
    int i = k * 256 + tid;
    float4 s = s4[i];
    int4   m = m4[i];
    float f0 = m.x ? 1.0f : 0.0f;
    float f1 = m.y ? 1.0f : 0.0f;
    float f2 = m.z ? 1.0f : 0.0f;
    float f3 = m.w ? 1.0f : 0.0f;
    aL = fmaf(f0 * s.x, s.x, aL);
    aL = fmaf(f1 * s.y, s.y, aL);
    aL = fmaf(f2 * s.z, s.z, aL);
    aL = fmaf(f3 * s.w, s.w, aL);
    aM += (f0 + f1) + (f2 + f3);
  }
  float sL = wave_sum_wmma(aL);
  float sM = wave_sum_wmma(aM);
  if ((tid & 31) == 0) {
    atomicAdd(&accLoss[b], sL);
    atomicAdd(&accMask[b], sM);
  }
}

// ---- K5: finalize --------------------------------------------------------
__global__ void k_final(const float* accLoss, const float* accDelta,
                        const float* accMask, float* out) {
  if (threadIdx.x == 0) {
    float acc = 0.0f;
    for (int b = 0; b < BATCH; ++b) {
      float denom = fmaxf(accMask[b], 1e-8f);
      acc += (accLoss[b] + accDelta[b]) / denom;
    }
    out[0] = acc / (float)BATCH;
  }
}

extern "C" void kernel_launch(void* const* d_in, const int* in_sizes, int n_in,
                              void* d_out, int out_size, void* d_ws,
                              size_t ws_size, hipStream_t stream) {
  const float* score1 = (const float*)d_in[0];
  const float* score2 = (const float*)d_in[1];
  const int*   mask   = (const int*)d_in[2];
  float* out = (float*)d_out;

  char* ws = (char*)d_ws;
  int*   candCount = (int*)(ws + OFF_CANDCNT);
  int*   kcount    = (int*)(ws + OFF_KCOUNT);
  float* accLoss   = (float*)(ws + OFF_ACCLOSS);
  float* accMask   = (float*)(ws + OFF_ACCMASK);
  float* accDelta  = (float*)(ws + OFF_ACCDELTA);
  int*   kp        = (int*)(ws + OFF_KP);
  unsigned long long* cand = (unsigned long long*)(ws + OFF_CAND);

  k_zero<<<1, 256, 0, stream>>>((unsigned*)d_ws);

  dim3 gNms(WW / TILE, HH / TILE, BATCH);
  k_nms_collect<<<gNms, 256, 0, stream>>>(score2, mask, candCount, cand);

  k_select<<<BATCH, 1024, 0, stream>>>(candCount, cand, kcount, kp);

  k_stamp<<<BATCH * TOPK, 64, 0, stream>>>(score1, mask, kcount, kp, accDelta);

  dim3 gBase(NPIX / 4096, BATCH);
  k_base<<<gBase, 256, 0, stream>>>(score1, mask, accLoss, accMask);

  k_final<<<1, 32, 0, stream>>>(accLoss, accDelta, accMask, out);
}